// HistogramLoss_52029233824305
// MI455X (gfx1250) — compile-verified
//
#include <hip/hip_runtime.h>

// HistogramLoss for MI455X (gfx1250, wave32).
// Kernel 1: streaming float4 loads of x and y, per-wave privatized LDS
//           histograms (ds_add_u32), global atomic merge.
// Kernel 2: KL(h_y || h_x) over 768 bins; final sum done with chained
//           v_wmma_f32_16x16x4_f32 (B = ones => layout-independent total).

#define WAVES_PER_BLOCK 8
#define HIST_PER_WAVE   1536   // 2 tensors * 3 channels * 256 bins
#define TOTAL_HIST      1536

typedef float v2f __attribute__((ext_vector_type(2)));
typedef float v8f __attribute__((ext_vector_type(8)));

__global__ void HistogramLoss_zero_ws(unsigned int* __restrict__ g) {
    int i = blockIdx.x * blockDim.x + threadIdx.x;
    if (i < TOTAL_HIST) g[i] = 0u;
}

__device__ __forceinline__ int binOf(float v) {
    // matches jnp: ((256-1) * clip(v,0,1)).astype(int32)  (truncation)
    return (int)(fminf(fmaxf(v, 0.0f), 1.0f) * 255.0f);
}

__global__ __launch_bounds__(256) void HistogramLoss_hist_kernel(
    const float* __restrict__ x, const float* __restrict__ y,
    unsigned int* __restrict__ ghist, int nv4) {
    // 48 KB: per-wave private histograms (wave32 -> tid>>5)
    __shared__ unsigned int lh[WAVES_PER_BLOCK * HIST_PER_WAVE];

    const int tid = threadIdx.x;
    unsigned int* wh = &lh[(tid >> 5) * HIST_PER_WAVE];

    for (int e = tid; e < WAVES_PER_BLOCK * HIST_PER_WAVE; e += 256)
        lh[e] = 0u;
    __syncthreads();

    const float4* __restrict__ xv = (const float4*)x;
    const float4* __restrict__ yv = (const float4*)y;
    const int stride = gridDim.x * 256;

    for (int v = blockIdx.x * 256 + tid; v < nv4; v += stride) {
        if (v + stride < nv4) {  // speculative prefetch of next tile -> global_prefetch_b8
            __builtin_prefetch(&xv[v + stride], 0, 0);
            __builtin_prefetch(&yv[v + stride], 0, 0);
        }
        // 262144 elements (65536 float4) per channel plane; planes cycle r,g,b
        const int ch = (v >> 16) % 3;
        unsigned int* hx = wh + ch * 256;        // tensor x histograms
        unsigned int* hy = wh + 768 + ch * 256;  // tensor y histograms

        const float4 a = xv[v];
        const float4 b = yv[v];
        atomicAdd(&hx[binOf(a.x)], 1u);   // ds_add_u32 (no rtn)
        atomicAdd(&hx[binOf(a.y)], 1u);
        atomicAdd(&hx[binOf(a.z)], 1u);
        atomicAdd(&hx[binOf(a.w)], 1u);
        atomicAdd(&hy[binOf(b.x)], 1u);
        atomicAdd(&hy[binOf(b.y)], 1u);
        atomicAdd(&hy[binOf(b.z)], 1u);
        atomicAdd(&hy[binOf(b.w)], 1u);
    }
    __syncthreads();

    // merge the 8 per-wave copies, then one global atomic per counter per block
    for (int e = tid; e < TOTAL_HIST; e += 256) {
        unsigned int s = 0;
#pragma unroll
        for (int w = 0; w < WAVES_PER_BLOCK; ++w)
            s += lh[w * HIST_PER_WAVE + e];
        atomicAdd(&ghist[e], s);  // global_atomic_add_u32
    }
}

// Single wave32. EXEC is all ones (no divergence before the WMMAs).
// total = sum of 768 KL terms. With B = all-ones, D[m][n] accumulates
// rowsum(A)[m] into every column, so sum_m D[m][0] = sum of all A entries --
// independent of the exact A VGPR layout.
__global__ __launch_bounds__(32) void HistogramLoss_kl_kernel(
    const unsigned int* __restrict__ gh, float* __restrict__ out) {
    const int lane = threadIdx.x;
    const float invN = 1.0f / 8388608.0f;  // 32*512*512 pixels per channel (exact)
    const float EPS = 1e-10f;

    v8f c = {0.f, 0.f, 0.f, 0.f, 0.f, 0.f, 0.f, 0.f};
    v2f ones;
    ones[0] = 1.0f;
    ones[1] = 1.0f;

#pragma unroll
    for (int i = 0; i < 12; ++i) {
        v2f a;
#pragma unroll
        for (int j = 0; j < 2; ++j) {
            const int g = i * 64 + lane * 2 + j;      // covers [0,768) exactly once
            const float q = (float)gh[g] * invN + EPS;        // h_x (predicted)
            const float p = (float)gh[768 + g] * invN + EPS;  // h_y (target)
            a[j] = p * logf(p / q);
        }
        c = __builtin_amdgcn_wmma_f32_16x16x4_f32(
            /*neg_a=*/false, a, /*neg_b=*/false, ones,
            /*c_mod=*/(short)0, c, /*reuse_a=*/false, /*reuse_b=*/false);
    }

    // D column 0 lives at N=0: lanes 0 (M=0..7 in vgprs 0..7) and 16 (M=8..15)
    float s = c[0] + c[1] + c[2] + c[3] + c[4] + c[5] + c[6] + c[7];
    float s2 = __shfl(s, (lane + 16) & 31, 32);
    if (lane == 0) out[0] = s + s2;
}

extern "C" void kernel_launch(void* const* d_in, const int* in_sizes, int n_in,
                              void* d_out, int out_size, void* d_ws, size_t ws_size,
                              hipStream_t stream) {
    const float* x = (const float*)d_in[0];
    const float* y = (const float*)d_in[1];
    unsigned int* ghist = (unsigned int*)d_ws;  // 1536 u32 = 6 KB scratch
    float* out = (float*)d_out;

    const int n = in_sizes[0];   // 25,165,824
    const int nv4 = n / 4;       // 6,291,456 float4 per tensor

    HistogramLoss_zero_ws<<<(TOTAL_HIST + 255) / 256, 256, 0, stream>>>(ghist);

    const int blocks = 2048;     // 16K waves; 12 float4-pairs per thread
    HistogramLoss_hist_kernel<<<blocks, 256, 0, stream>>>(x, y, ghist, nv4);

    HistogramLoss_kl_kernel<<<1, 32, 0, stream>>>(ghist, out);
}